// MarginalCalibrationError_46188078301368
// MI455X (gfx1250) — compile-verified
//
#include <hip/hip_runtime.h>
#include <stdint.h>
#include <stddef.h>

#define NBINS   10
#define THREADS 256
#define VEC     4
#define TILE    (THREADS * VEC)   // 1024 floats = 4 KB per stage buffer
#define DEPTH   4                 // async pipeline depth (buffers)
#define MAXC    128
#define GRID    2048

// searchsorted(linspace(0,1,NBINS+1), p, side='left') - 1 for p in (0,1].
// Edges computed as i*0.1f in fp32 to match jnp.linspace (iota * step in f32).
__device__ __forceinline__ int bin_of(float p) {
    int b = 0;
#pragma unroll
    for (int i = 1; i <= NBINS; ++i)
        b += (p > (float)i * 0.1f) ? 1 : 0;
    return b;
}

__global__ void zero_ws_kernel(float* ws, int nfloats) {
    int i = blockIdx.x * blockDim.x + threadIdx.x;
    if (i < nfloats) ws[i] = 0.0f;
}

__global__ __launch_bounds__(THREADS)
void mce_accum_kernel(const float* __restrict__ probas,
                      const int*   __restrict__ labels,
                      float* __restrict__ ghist,          // [3][c*NBINS]: cnt, sum_p, sum_acc
                      unsigned long long total, int c,
                      unsigned long long numFull)
{
    __shared__ float stage[DEPTH][TILE];
    __shared__ float s_cnt[MAXC * NBINS];
    __shared__ float s_sum[MAXC * NBINS];
    __shared__ float s_acc[MAXC * NBINS];

    const int  tid    = threadIdx.x;
    const int  slots  = c * NBINS;
    const bool useLds = (c <= MAXC);
    float* g_cnt = ghist;
    float* g_sum = ghist + slots;
    float* g_acc = ghist + 2 * slots;

    if (useLds) {
        for (int i = tid; i < slots; i += THREADS) {
            s_cnt[i] = 0.0f; s_sum[i] = 0.0f; s_acc[i] = 0.0f;
        }
        __syncthreads();
    }

    // per-lane async issue: each lane stages its own 16B (ASYNCcnt path)
    auto issue = [&](unsigned long long tt, int buf) {
        const float* gp = probas + tt * TILE + (unsigned long long)tid * VEC;
        unsigned lds = (unsigned)(uintptr_t)&stage[buf][tid * VEC];
        unsigned long long ga = (unsigned long long)(uintptr_t)gp;
        asm volatile("global_load_async_to_lds_b128 %0, %1, off"
                     :: "v"(lds), "v"(ga) : "memory");
    };

    const unsigned long long t0        = blockIdx.x;
    const unsigned long long stepTiles = gridDim.x;

    // ---- prologue: fill pipeline up to DEPTH-1 tiles ahead ----
    int outstanding = 0;
#pragma unroll
    for (int d = 0; d < DEPTH - 1; ++d) {
        unsigned long long tt = t0 + (unsigned long long)d * stepTiles;
        if (tt < numFull) { issue(tt, d); ++outstanding; }
    }

    // ---- incremental (row, col) tracking: divisions only once per thread ----
    unsigned long long nrow = 0; int cc = 0;
    unsigned long long step_q = 0; int step_r = 0;
    if (t0 < numFull) {
        unsigned long long base0 = t0 * TILE + (unsigned long long)tid * VEC;
        nrow = base0 / (unsigned)c;
        cc   = (int)(base0 - nrow * (unsigned)c);
        unsigned long long stepElems = stepTiles * (unsigned long long)TILE;
        step_q = stepElems / (unsigned)c;
        step_r = (int)(stepElems - step_q * (unsigned)c);
    }

    unsigned k = 0;
    for (unsigned long long t = t0; t < numFull; t += stepTiles, ++k) {
        unsigned long long nxt = t + (unsigned long long)(DEPTH - 1) * stepTiles;
        if (nxt < numFull) { issue(nxt, (int)((k + DEPTH - 1) % DEPTH)); ++outstanding; }

        // in-order async completion: outstanding-1 left => oldest (tile t) landed
        switch (outstanding) {
            case 4:  asm volatile("s_wait_asynccnt 3" ::: "memory"); break;
            case 3:  asm volatile("s_wait_asynccnt 2" ::: "memory"); break;
            case 2:  asm volatile("s_wait_asynccnt 1" ::: "memory"); break;
            default: asm volatile("s_wait_asynccnt 0" ::: "memory"); break;
        }

        // consume tile t from stage[k % DEPTH] (each lane reads only its own 16B)
        float4 v = *(const float4*)&stage[k % DEPTH][tid * VEC];
        unsigned long long rn = nrow; int rc = cc;
        int lab = labels[rn];
        float pv[VEC] = {v.x, v.y, v.z, v.w};
#pragma unroll
        for (int j = 0; j < VEC; ++j) {
            float p = pv[j];
            if (p > 0.0f) {                              // p==0 belongs to no bin
                int slot = rc * NBINS + bin_of(p);
                bool corr = (lab == rc);
                if (useLds) {
                    atomicAdd(&s_cnt[slot], 1.0f);
                    atomicAdd(&s_sum[slot], p);
                    if (corr) atomicAdd(&s_acc[slot], 1.0f);
                } else {
                    atomicAdd(&g_cnt[slot], 1.0f);
                    atomicAdd(&g_sum[slot], p);
                    if (corr) atomicAdd(&g_acc[slot], 1.0f);
                }
            }
            if (j < VEC - 1) {
                if (++rc == c) { rc = 0; ++rn; lab = labels[rn]; }
            }
        }
        --outstanding;

        // advance (row, col) by the uniform per-iteration stride
        cc += step_r;
        if (cc >= c) { cc -= c; nrow += 1; }
        nrow += step_q;
    }

    // ---- partial tail tile (at most one in the whole grid) ----
    unsigned long long tailStart = numFull * (unsigned long long)TILE;
    if (tailStart < total && blockIdx.x == (unsigned)(numFull % gridDim.x)) {
        for (unsigned long long g = tailStart + tid; g < total; g += THREADS) {
            float p = probas[g];
            if (p > 0.0f) {
                unsigned long long rn = g / (unsigned)c;
                int rc = (int)(g - rn * (unsigned)c);
                int slot = rc * NBINS + bin_of(p);
                bool corr = (labels[rn] == rc);
                if (useLds) {
                    atomicAdd(&s_cnt[slot], 1.0f);
                    atomicAdd(&s_sum[slot], p);
                    if (corr) atomicAdd(&s_acc[slot], 1.0f);
                } else {
                    atomicAdd(&g_cnt[slot], 1.0f);
                    atomicAdd(&g_sum[slot], p);
                    if (corr) atomicAdd(&g_acc[slot], 1.0f);
                }
            }
        }
    }

    // ---- flush privatized histogram ----
    if (useLds) {
        __syncthreads();
        for (int i = tid; i < slots; i += THREADS) {
            float a = s_cnt[i], b = s_sum[i], d = s_acc[i];
            if (a != 0.0f) atomicAdd(&g_cnt[i], a);
            if (b != 0.0f) atomicAdd(&g_sum[i], b);
            if (d != 0.0f) atomicAdd(&g_acc[i], d);
        }
    }
}

__global__ __launch_bounds__(128)
void mce_final_kernel(const float* __restrict__ ghist, float* __restrict__ out, int c)
{
    __shared__ float red[128];
    const float* g_cnt = ghist;
    const float* g_sum = ghist + c * NBINS;
    const float* g_acc = ghist + 2 * c * NBINS;

    float sq = 0.0f;
    for (int cc = threadIdx.x; cc < c; cc += blockDim.x) {
        float tot = 0.0f;
        for (int b = 0; b < NBINS; ++b) tot += g_cnt[cc * NBINS + b];
        if (tot > 0.0f) {
            for (int b = 0; b < NBINS; ++b) {
                float nb = g_cnt[cc * NBINS + b];
                if (nb > 0.0f) {
                    float conf = g_sum[cc * NBINS + b] / nb;
                    float acc  = g_acc[cc * NBINS + b] / nb;
                    float d = conf - acc;
                    sq += (nb / tot) * d * d;
                }
            }
        }
    }
    red[threadIdx.x] = sq;
    __syncthreads();
    for (int off = 64; off > 0; off >>= 1) {
        if (threadIdx.x < (unsigned)off) red[threadIdx.x] += red[threadIdx.x + off];
        __syncthreads();
    }
    if (threadIdx.x == 0) out[0] = sqrtf(red[0] / (float)c);
}

extern "C" void kernel_launch(void* const* d_in, const int* in_sizes, int n_in,
                              void* d_out, int out_size, void* d_ws, size_t ws_size,
                              hipStream_t stream) {
    const float* probas = (const float*)d_in[0];
    const int*   labels = (const int*)d_in[1];
    unsigned long long total = (unsigned long long)in_sizes[0];   // N*C
    int n = in_sizes[1];                                          // N
    int c = (n > 0) ? (int)(total / (unsigned long long)n) : 1;   // C (=80)
    int slots = c * NBINS;

    float* ws = (float*)d_ws;   // [3 * slots] fp32: cnt | sum_p | sum_acc
    int zf = 3 * slots;
    zero_ws_kernel<<<(zf + 255) / 256, 256, 0, stream>>>(ws, zf);

    unsigned long long numFull = total / TILE;
    int grid = GRID;
    if (numFull < (unsigned long long)grid) grid = (int)(numFull ? numFull : 1);

    mce_accum_kernel<<<grid, THREADS, 0, stream>>>(probas, labels, ws, total, c, numFull);
    mce_final_kernel<<<1, 128, 0, stream>>>(ws, (float*)d_out, c);
}